// Denoising_Unet_22978075033939
// MI455X (gfx1250) — compile-verified
//
#include <hip/hip_runtime.h>
#include <math.h>

// ---------------------------------------------------------------------------
// Denoising GATv2-UNet for MI455X (gfx1250, wave32).
// fp32 end-to-end; GEMMs use V_WMMA_F32_16X16X4_F32 (full-precision matrix op).
// ---------------------------------------------------------------------------

#define NNODES 50000
#define NEDGES 500000
#define BGRAPH 64
#define HID    128
#define NHEAD  8

typedef float v2f __attribute__((ext_vector_type(2)));
typedef float v8f __attribute__((ext_vector_type(8)));

__device__ __forceinline__ float gelu_f(float x) {
  return 0.5f * x * (1.0f + erff(x * 0.70710678118654752440f));
}
__device__ __forceinline__ float silu_f(float x) {
  return x / (1.0f + expf(-x));
}
__device__ __forceinline__ float lrelu_f(float x) {
  return x > 0.0f ? x : 0.2f * x;
}

// ---------------------------------------------------------------------------
// WMMA fp32 GEMM:  out[M,NOUT] = act( A[M,K] @ W[K,NOUT] + bias ) (+ res)
// Block: 256 threads = 8 wave32s; wave w owns rows [blk*128+16w, +16),
// grid.y tiles NOUT in chunks of 128 columns.
//
// Weights staged in LDS in K-PAIR INTERLEAVED layout:
//   sW[kpair * LSTRIDE + c*2 + 0] = W[2*kpair + 0][c]
//   sW[kpair * LSTRIDE + c*2 + 1] = W[2*kpair + 1][c]
// so each B fragment (rows k,k+1 at column c) is one aligned ds_load_b64 into
// an even VGPR pair (no v_mov assembly). LSTRIDE = 2*NB + 32 floats makes the
// half-wave (half=1, kpair+1) land on the opposite 32 LDS banks -> conflict-
// free b64 reads across the wave.
//
// A-frag (16x4 f32): lane l in [0,16): (M=l, K=k0..k0+1); l in [16,32):
//   (M=l-16, K=k0+2..k0+3). All 8 A-frags of a K=32 chunk are preloaded into
//   registers (8 back-to-back global_load_b64) before the WMMA loop.
// C/D (16x16 f32, 8 VGPRs): vgpr r, lane l -> row r + (l>>4)*8, col l&15.
//
// AMODE: 0 = plain A;  1 = A is concat(x_t[row,0:128], t_emb[gid[row],0:128])
// ACT:   0 = none; 1 = exact gelu
// RES:   1 = add res[row,col] after activation
// ---------------------------------------------------------------------------
template<int K, int NOUT, int AMODE, int ACT, int RES>
__global__ __launch_bounds__(256) void gemm_wmma_k(
    const float* __restrict__ A, const float* __restrict__ W,
    const float* __restrict__ bias, const float* __restrict__ res,
    float* __restrict__ out, int M,
    const float* __restrict__ A2, const int* __restrict__ gids)
{
  constexpr int KT = 32;                // K chunk staged in LDS
  constexpr int NB = 128;               // columns handled per block
  constexpr int LSTRIDE = 2 * NB + 32;  // floats; bank-conflict-free padding
  __shared__ float sW[(KT / 2) * LSTRIDE];  // 18KB

  const int tid  = threadIdx.x;
  const int wave = tid >> 5;
  const int lane = tid & 31;
  const int half = lane >> 4;  // 0 or 1
  const int l15  = lane & 15;
  const int rowbase = blockIdx.x * 128 + wave * 16;
  const int colbase = blockIdx.y * NB;

  int arow = rowbase + l15;
  if (arow >= M) arow = M - 1;  // clamp so loads stay in-bounds (stores guarded)
  int gid = 0;
  if (AMODE == 1) gid = gids[arow];

  v8f acc[NB / 16] = {};

  for (int kb = 0; kb < K; kb += KT) {
    // --- stage W[kb..kb+31][colbase..+127] -> LDS, k-pair interleaved ---
    // j indexes (kpair, cpair): (KT/2)*(NB/2) = 1024 float4 stores, 4/thread.
    for (int j = tid; j < (KT / 2) * (NB / 2); j += 256) {
      const int kp = j >> 6;            // 0..15
      const int cc = (j & 63) * 2;      // 0,2,..,126
      const float2 r0 =
          *(const float2*)(W + (size_t)(kb + kp * 2 + 0) * NOUT + colbase + cc);
      const float2 r1 =
          *(const float2*)(W + (size_t)(kb + kp * 2 + 1) * NOUT + colbase + cc);
      const float4 st = {r0.x, r1.x, r0.y, r1.y};
      *(float4*)(sW + kp * LSTRIDE + cc * 2) = st;
    }
    __syncthreads();

    // --- preload all 8 A fragments of this chunk (hides global latency) ---
    v2f afr[KT / 4];
    #pragma unroll
    for (int k0 = 0; k0 < KT; k0 += 4) {
      const int kk = kb + k0 + half * 2;
      if (AMODE == 0) {
        const float2 t2 = *(const float2*)(A + (size_t)arow * K + kk);
        afr[k0 / 4].x = t2.x; afr[k0 / 4].y = t2.y;
      } else {
        const float2 t2 = (kk < HID)
            ? *(const float2*)(A  + (size_t)arow * HID + kk)
            : *(const float2*)(A2 + (size_t)gid  * HID + (kk - HID));
        afr[k0 / 4].x = t2.x; afr[k0 / 4].y = t2.y;
      }
    }

    // --- WMMA loop: one ds_load_b64 per B frag, 8 wmma per K-step ---
    #pragma unroll
    for (int k0 = 0; k0 < KT; k0 += 4) {
      const int kpair = k0 / 2 + half;
      const v2f a = afr[k0 / 4];
      #pragma unroll
      for (int t = 0; t < NB / 16; ++t) {
        const int c = t * 16 + l15;
        const v2f b = *(const v2f*)(sW + kpair * LSTRIDE + c * 2);
        acc[t] = __builtin_amdgcn_wmma_f32_16x16x4_f32(
            false, a, false, b, (short)0, acc[t], false, false);
      }
    }
    __syncthreads();
  }

  #pragma unroll
  for (int t = 0; t < NB / 16; ++t) {
    const int c  = colbase + t * 16 + l15;
    const float bv = bias[c];
    #pragma unroll
    for (int r = 0; r < 8; ++r) {
      const int grow = rowbase + r + half * 8;
      if (grow < M) {
        float v = acc[t][r] + bv;
        if (ACT == 1) v = gelu_f(v);
        if (RES == 1) v += res[(size_t)grow * NOUT + c];
        out[(size_t)grow * NOUT + c] = v;
      }
    }
  }
}

// ---------------------------------------------------------------------------
// LayerNorm over HID=128: one wave32 per row, float4 per lane, shfl_xor tree.
// ---------------------------------------------------------------------------
__global__ __launch_bounds__(256) void layernorm_k(
    const float* __restrict__ x, const float* __restrict__ g,
    const float* __restrict__ b, float* __restrict__ y, int M)
{
  const int wave = threadIdx.x >> 5;
  const int lane = threadIdx.x & 31;
  const int row  = blockIdx.x * 8 + wave;
  if (row >= M) return;
  float4 v = *(const float4*)(x + (size_t)row * HID + lane * 4);
  float s = v.x + v.y + v.z + v.w;
  #pragma unroll
  for (int o = 16; o > 0; o >>= 1) s += __shfl_xor(s, o, 32);
  const float mu = s * (1.0f / HID);
  float4 d = {v.x - mu, v.y - mu, v.z - mu, v.w - mu};
  float q = d.x * d.x + d.y * d.y + d.z * d.z + d.w * d.w;
  #pragma unroll
  for (int o = 16; o > 0; o >>= 1) q += __shfl_xor(q, o, 32);
  const float rs = rsqrtf(q * (1.0f / HID) + 1e-5f);
  const float4 gg = *(const float4*)(g + lane * 4);
  const float4 bb = *(const float4*)(b + lane * 4);
  float4 o4 = {d.x * rs * gg.x + bb.x, d.y * rs * gg.y + bb.y,
               d.z * rs * gg.z + bb.z, d.w * rs * gg.w + bb.w};
  *(float4*)(y + (size_t)row * HID + lane * 4) = o4;
}

// ---------------------------------------------------------------------------
// GATv2 edge kernels (all L2-resident scatter/gather; fp32 global atomics).
// ---------------------------------------------------------------------------
__global__ void init_gat_k(float* __restrict__ gat, float* __restrict__ m,
                           float* __restrict__ z, int n)
{
  const int idx = blockIdx.x * blockDim.x + threadIdx.x;  // over N*HID
  if (idx < n * HID) gat[idx] = 0.0f;
  if (idx < n * NHEAD) { m[idx] = -INFINITY; z[idx] = 0.0f; }
}

// score[e,h] = sum_d attn[h,d] * leaky_relu(fs[src,h,d] + fd[dst,h,d]);
// atomicMax into m[dst,h]
__global__ void edge_score_k(const float* __restrict__ fs,
                             const float* __restrict__ fd,
                             const float* __restrict__ attn,
                             const int* __restrict__ src,
                             const int* __restrict__ dst,
                             float* __restrict__ score,
                             float* __restrict__ m, int E)
{
  const int idx = blockIdx.x * blockDim.x + threadIdx.x;  // (e,h)
  if (idx >= E * NHEAD) return;
  const int e = idx >> 3, h = idx & 7;
  const int s = src[e], d = dst[e];
  const float4* pf = (const float4*)(fs + (size_t)s * HID + h * 16);
  const float4* pd = (const float4*)(fd + (size_t)d * HID + h * 16);
  const float4* pa = (const float4*)(attn + h * 16);
  float sc = 0.0f;
  #pragma unroll
  for (int q = 0; q < 4; ++q) {
    const float4 a = pf[q], b = pd[q], w = pa[q];
    sc += w.x * lrelu_f(a.x + b.x) + w.y * lrelu_f(a.y + b.y) +
          w.z * lrelu_f(a.z + b.z) + w.w * lrelu_f(a.w + b.w);
  }
  score[idx] = sc;
  atomicMax(&m[(size_t)d * NHEAD + h], sc);
}

__global__ void edge_exp_k(float* __restrict__ score, const float* __restrict__ m,
                           float* __restrict__ z, const int* __restrict__ dst, int E)
{
  const int idx = blockIdx.x * blockDim.x + threadIdx.x;
  if (idx >= E * NHEAD) return;
  const int e = idx >> 3, h = idx & 7;
  const int d = dst[e];
  const float a = expf(score[idx] - m[(size_t)d * NHEAD + h]);
  score[idx] = a;
  atomicAdd(&z[(size_t)d * NHEAD + h], a);
}

__global__ void edge_norm_k(float* __restrict__ score, const float* __restrict__ z,
                            const int* __restrict__ dst, int E)
{
  const int idx = blockIdx.x * blockDim.x + threadIdx.x;
  if (idx >= E * NHEAD) return;
  const int e = idx >> 3, h = idx & 7;
  score[idx] = score[idx] / z[(size_t)dst[e] * NHEAD + h];
}

// gat[dst, c] += a[e, c>>4] * fs[src, c]
__global__ void edge_agg_k(const float* __restrict__ score,
                           const float* __restrict__ fs,
                           const int* __restrict__ src,
                           const int* __restrict__ dst,
                           float* __restrict__ gat, int E)
{
  const int idx = blockIdx.x * blockDim.x + threadIdx.x;  // (e, c)
  if (idx >= E * HID) return;
  const int e = idx >> 7, c = idx & 127, h = c >> 4;
  const float w = score[(size_t)e * NHEAD + h];
  const float v = w * fs[(size_t)src[e] * HID + c];
  atomicAdd(&gat[(size_t)dst[e] * HID + c], v);
}

// hout = gelu(gat) + tc[gid[n]] + hin   (residual + time conditioning)
__global__ void combine_k(const float* __restrict__ gat, const float* __restrict__ tc,
                          const int* __restrict__ gids, const float* __restrict__ hin,
                          float* __restrict__ hout, int n)
{
  const int idx = blockIdx.x * blockDim.x + threadIdx.x;
  if (idx >= n * HID) return;
  const int nn = idx >> 7, c = idx & 127;
  hout[idx] = gelu_f(gat[idx]) + tc[(size_t)gids[nn] * HID + c] + hin[idx];
}

__global__ void add_inplace_k(float* __restrict__ y, const float* __restrict__ s, int n)
{
  const int idx = blockIdx.x * blockDim.x + threadIdx.x;
  if (idx < n) y[idx] += s[idx];
}

__global__ void copy_k(const float* __restrict__ x, float* __restrict__ y, int n)
{
  const int idx = blockIdx.x * blockDim.x + threadIdx.x;
  if (idx < n) y[idx] = x[idx];
}

// ---------------------------------------------------------------------------
// Tiny dense (B=64 rows): out[r,j] = act( sum_k pre(in[r,k]) * W[k,j] + b[j] )
// One block per row; input row staged in LDS.
// PRE: 1 = silu on input.  ACT: 1 = gelu on output.
// ---------------------------------------------------------------------------
template<int K, int NOUT, int PRE, int ACT>
__global__ void dense_small_k(const float* __restrict__ in, const float* __restrict__ W,
                              const float* __restrict__ bias, float* __restrict__ out)
{
  __shared__ float si[K];
  const int r = blockIdx.x;
  for (int k = threadIdx.x; k < K; k += blockDim.x) {
    float v = in[(size_t)r * K + k];
    if (PRE == 1) v = silu_f(v);
    si[k] = v;
  }
  __syncthreads();
  const int j = threadIdx.x;
  if (j < NOUT) {
    float s = bias[j];
    for (int k = 0; k < K; ++k) s += si[k] * W[(size_t)k * NOUT + j];
    if (ACT == 1) s = gelu_f(s);
    out[(size_t)r * NOUT + j] = s;
  }
}

// Sinusoidal time embedding: emb[b, j<64]=sin(tf*freq(j)), j>=64 -> cos.
__global__ void time_emb0_k(const int* __restrict__ t, float* __restrict__ emb)
{
  const int b = blockIdx.x, j = threadIdx.x;  // 128 threads
  const float tf = (float)t[b] * (1.0f / 1000.0f);
  const int i = j & 63;
  const float freq = expf((float)i * (-logf(10000.0f) / 63.0f));
  const float v = tf * freq;
  emb[(size_t)b * HID + j] = (j < 64) ? sinf(v) : cosf(v);
}

// ---------------------------------------------------------------------------
// Host side
// ---------------------------------------------------------------------------
namespace {

struct BlkP {
  const float *ln1_g, *ln1_b, *ln2_g, *ln2_b;
  const float *Wsrc, *bsrc, *Wdst, *bdst, *attn, *Wt, *bt, *Wf1, *bf1, *Wf2, *bf2;
};

struct Ws {
  float *xn, *fs, *fd, *gat, *ffn, *m, *z, *a, *tc, *t_emb;
};

constexpr int GEMM_GX = (NNODES + 127) / 128;          // 391
constexpr int LN_G    = (NNODES + 7) / 8;              // 6250
constexpr int NW_G    = (NNODES * HID + 255) / 256;    // 25000
constexpr int EH_G    = (NEDGES * NHEAD + 255) / 256;  // 15625
constexpr int EA_G    = (NEDGES * HID + 255) / 256;    // 250000

void run_block(const BlkP& p, const float* hin, float* hout, const Ws& w,
               const int* e_src, const int* e_dst, const int* gids,
               hipStream_t stream)
{
  // GATv2 on ln1(hin)
  layernorm_k<<<LN_G, 256, 0, stream>>>(hin, p.ln1_g, p.ln1_b, w.xn, NNODES);
  gemm_wmma_k<128, 128, 0, 0, 0><<<dim3(GEMM_GX, 1), 256, 0, stream>>>(
      w.xn, p.Wsrc, p.bsrc, nullptr, w.fs, NNODES, nullptr, nullptr);
  gemm_wmma_k<128, 128, 0, 0, 0><<<dim3(GEMM_GX, 1), 256, 0, stream>>>(
      w.xn, p.Wdst, p.bdst, nullptr, w.fd, NNODES, nullptr, nullptr);
  init_gat_k<<<NW_G, 256, 0, stream>>>(w.gat, w.m, w.z, NNODES);
  edge_score_k<<<EH_G, 256, 0, stream>>>(w.fs, w.fd, p.attn, e_src, e_dst,
                                         w.a, w.m, NEDGES);
  edge_exp_k<<<EH_G, 256, 0, stream>>>(w.a, w.m, w.z, e_dst, NEDGES);
  edge_norm_k<<<EH_G, 256, 0, stream>>>(w.a, w.z, e_dst, NEDGES);
  edge_agg_k<<<EA_G, 256, 0, stream>>>(w.a, w.fs, e_src, e_dst, w.gat, NEDGES);
  // time conditioning + residual
  dense_small_k<128, 128, 0, 1><<<BGRAPH, 128, 0, stream>>>(w.t_emb, p.Wt, p.bt, w.tc);
  combine_k<<<NW_G, 256, 0, stream>>>(w.gat, w.tc, gids, hin, hout, NNODES);
  // FFN with residual
  layernorm_k<<<LN_G, 256, 0, stream>>>(hout, p.ln2_g, p.ln2_b, w.xn, NNODES);
  gemm_wmma_k<128, 256, 0, 1, 0><<<dim3(GEMM_GX, 2), 256, 0, stream>>>(
      w.xn, p.Wf1, p.bf1, nullptr, w.ffn, NNODES, nullptr, nullptr);
  gemm_wmma_k<256, 128, 0, 0, 1><<<dim3(GEMM_GX, 1), 256, 0, stream>>>(
      w.ffn, p.Wf2, p.bf2, hout, hout, NNODES, nullptr, nullptr);
}

}  // namespace

extern "C" void kernel_launch(void* const* d_in, const int* in_sizes, int n_in,
                              void* d_out, int out_size, void* d_ws, size_t ws_size,
                              hipStream_t stream)
{
  (void)in_sizes; (void)n_in; (void)out_size; (void)ws_size;
  // Param pytree flattened in setup_inputs() insertion order:
  //  0..5   time:   W1[128,256] b1[256] W2[256,128] b2[128] Wm[128,128] bm[128]
  //  6..7   input_proj: W[256,128] b[128]
  //  8..97  blocks[6] x {ln1_g,ln1_b,ln2_g,ln2_b,Wsrc,bsrc,Wdst,bdst,attn,
  //                      Wt,bt,Wf1,bf1,Wf2,bf2}
  //  98..103 output: ln_g ln_b W1[128,128] b1 W2[128,128] b2
  //  104 x_t[50000,128] 105 time_indices[64] 106 edge_src 107 edge_dst
  //  108 node_graph_ids[50000]
  #define PF(i) ((const float*)d_in[(i)])
  const float* tW1 = PF(0);  const float* tb1 = PF(1);
  const float* tW2 = PF(2);  const float* tb2 = PF(3);
  const float* tWm = PF(4);  const float* tbm = PF(5);
  const float* ipW = PF(6);  const float* ipb = PF(7);
  const float* o_lng = PF(98); const float* o_lnb = PF(99);
  const float* oW1 = PF(100); const float* ob1 = PF(101);
  const float* oW2 = PF(102); const float* ob2 = PF(103);
  const float* x_t  = PF(104);
  const int* t_idx  = (const int*)d_in[105];
  const int* e_src  = (const int*)d_in[106];
  const int* e_dst  = (const int*)d_in[107];
  const int* gids   = (const int*)d_in[108];

  BlkP blk[6];
  for (int k = 0; k < 6; ++k) {
    const int b = 8 + 15 * k;
    blk[k] = {PF(b + 0), PF(b + 1), PF(b + 2),  PF(b + 3),  PF(b + 4),
              PF(b + 5), PF(b + 6), PF(b + 7),  PF(b + 8),  PF(b + 9),
              PF(b + 10), PF(b + 11), PF(b + 12), PF(b + 13), PF(b + 14)};
  }
  #undef PF

  // Workspace arena (fp32)
  float* ws = (float*)d_ws;
  size_t o = 0;
  float* temb0 = ws + o; o += (size_t)BGRAPH * HID;
  float* tbA   = ws + o; o += (size_t)BGRAPH * 2 * HID;
  float* tbB   = ws + o; o += (size_t)BGRAPH * HID;
  Ws w;
  w.t_emb = ws + o; o += (size_t)BGRAPH * HID;
  w.tc    = ws + o; o += (size_t)BGRAPH * HID;
  float* s0 = ws + o; o += (size_t)NNODES * HID;
  float* s1 = ws + o; o += (size_t)NNODES * HID;
  float* s2 = ws + o; o += (size_t)NNODES * HID;
  float* hA = ws + o; o += (size_t)NNODES * HID;
  float* hB = ws + o; o += (size_t)NNODES * HID;
  w.xn  = ws + o; o += (size_t)NNODES * HID;
  w.fs  = ws + o; o += (size_t)NNODES * HID;
  w.fd  = ws + o; o += (size_t)NNODES * HID;
  w.gat = ws + o; o += (size_t)NNODES * HID;
  w.ffn = ws + o; o += (size_t)NNODES * 2 * HID;
  w.m   = ws + o; o += (size_t)NNODES * NHEAD;
  w.z   = ws + o; o += (size_t)NNODES * NHEAD;
  w.a   = ws + o; o += (size_t)NEDGES * NHEAD;

  // --- time embedding: sin/cos -> MLP -> silu -> Wm ---
  time_emb0_k<<<BGRAPH, HID, 0, stream>>>(t_idx, temb0);
  dense_small_k<128, 256, 0, 1><<<BGRAPH, 256, 0, stream>>>(temb0, tW1, tb1, tbA);
  dense_small_k<256, 128, 0, 0><<<BGRAPH, 128, 0, stream>>>(tbA, tW2, tb2, tbB);
  dense_small_k<128, 128, 1, 0><<<BGRAPH, 128, 0, stream>>>(tbB, tWm, tbm, w.t_emb);

  // --- input projection: gelu(concat(x_t, t_emb[gids]) @ ipW + ipb) -> s0 ---
  gemm_wmma_k<256, 128, 1, 1, 0><<<dim3(GEMM_GX, 1), 256, 0, stream>>>(
      x_t, ipW, ipb, nullptr, s0, NNODES, w.t_emb, gids);

  // --- UNet: down(2) / middle / up(2) / final with skip connections ---
  run_block(blk[0], s0, s1, w, e_src, e_dst, gids, stream);  // h1 (skip)
  run_block(blk[1], s1, s2, w, e_src, e_dst, gids, stream);  // h2 (skip)
  run_block(blk[2], s2, hA, w, e_src, e_dst, gids, stream);  // middle
  add_inplace_k<<<NW_G, 256, 0, stream>>>(hA, s2, NNODES * HID);
  run_block(blk[3], hA, hB, w, e_src, e_dst, gids, stream);
  add_inplace_k<<<NW_G, 256, 0, stream>>>(hB, s1, NNODES * HID);
  run_block(blk[4], hB, hA, w, e_src, e_dst, gids, stream);
  add_inplace_k<<<NW_G, 256, 0, stream>>>(hA, s0, NNODES * HID);
  run_block(blk[5], hA, hB, w, e_src, e_dst, gids, stream);  // h (2nd output)

  // --- output head: denoised = gelu(ln(h)@W1+b1)@W2+b2 ---
  float* dout = (float*)d_out;
  layernorm_k<<<LN_G, 256, 0, stream>>>(hB, o_lng, o_lnb, w.xn, NNODES);
  gemm_wmma_k<128, 128, 0, 1, 0><<<dim3(GEMM_GX, 1), 256, 0, stream>>>(
      w.xn, oW1, ob1, nullptr, w.gat, NNODES, nullptr, nullptr);
  gemm_wmma_k<128, 128, 0, 0, 0><<<dim3(GEMM_GX, 1), 256, 0, stream>>>(
      w.gat, oW2, ob2, nullptr, dout, NNODES, nullptr, nullptr);
  copy_k<<<NW_G, 256, 0, stream>>>(hB, dout + (size_t)NNODES * HID, NNODES * HID);
}